// RWKVRegionCell_7275674599972
// MI455X (gfx1250) — compile-verified
//
#include <hip/hip_runtime.h>
#include <hip/hip_bf16.h>
#include <math.h>

// ---------------------------------------------------------------------------
// Problem constants (from reference)
// ---------------------------------------------------------------------------
#define D 8192
#define M_TIME 16
#define KWTA_K (D / 8)
#define EPS_DIV 1e-6f
#define MAX_EXP 30.0f
#define MAX_NOISE 10000.0f
#define RMS_EPS 1e-6f

// GEMV tiling
#define KT 32          // K-chunk staged in LDS per iteration
#define KTP 36         // padded LDS row stride (x4B: multiple of 16B for b128,
                       // stride-36 banks -> conflict-free 16-lane column reads)
#define RPB 64         // rows per block (4 waves * 16 rows)
#define TPB 128        // threads per block (4 wave32)
#define NTILES (D / KT)
// async b128 copies issued per wave per tile: (RPB*KT/4) chunks / TPB threads
#define ASYNC_PER_TILE ((RPB * KT / 4) / TPB)   // = 4

typedef __attribute__((ext_vector_type(2))) float v2f;
typedef __attribute__((ext_vector_type(8))) float v8f;

// ---------------------------------------------------------------------------
// Kernel 1: x = rmsnorm(x_in + pred) * g_norm   (single block)
// ---------------------------------------------------------------------------
__global__ void rmsnorm_kernel(const float* __restrict__ x_in,
                               const float* __restrict__ pred,
                               const float* __restrict__ g,
                               float* __restrict__ xout) {
  __shared__ float red[256];
  const int t = threadIdx.x;
  float ss = 0.f;
  for (int i = t; i < D; i += 256) {
    float s = x_in[i] + pred[i];
    xout[i] = s;
    ss += s * s;
  }
  red[t] = ss;
  __syncthreads();
  for (int off = 128; off > 0; off >>= 1) {
    if (t < off) red[t] += red[t + off];
    __syncthreads();
  }
  const float scale = rsqrtf(red[0] / (float)D + RMS_EPS);
  for (int i = t; i < D; i += 256) {
    xout[i] = xout[i] * scale * g[i];
  }
}

// ---------------------------------------------------------------------------
// Async global->LDS staging of one RPB x KT weight tile (CDNA5 async path,
// tracked by ASYNCcnt; bypasses the VGPR round-trip entirely).
// Manually unrolled with structurally in-bounds indices so the compiler emits
// 4 back-to-back async issues per wave with NO exec-mask manipulation.
// 8 consecutive threads cover one 128B row segment -> perfect coalescing.
// ---------------------------------------------------------------------------
__device__ __forceinline__ void stage_tile_async(const float* __restrict__ W,
                                                 int rowTileBase, int k0,
                                                 float (*buf)[KTP]) {
  const int t = threadIdx.x;
#pragma unroll
  for (int j = 0; j < ASYNC_PER_TILE; ++j) {
    const int i = t + j * TPB;       // always < RPB*KT/4 = 512: no bounds check
    const int r = i >> 3;            // KT/4 = 8 float4 chunks per row
    const int cc = (i & 7) * 4;
    const unsigned long long gptr =
        (unsigned long long)(W + (size_t)(rowTileBase + r) * D + k0 + cc);
    // Flat shared pointer low 32 bits == LDS byte offset (aperture rule).
    const unsigned lptr = (unsigned)(unsigned long long)&buf[r][cc];  // 16B aligned
    asm volatile("global_load_async_to_lds_b128 %0, %1, off"
                 :: "v"(lptr), "v"(gptr)
                 : "memory");
  }
}

__device__ __forceinline__ void wait_async_n() {
  asm volatile("s_wait_asynccnt 0x4" ::: "memory");  // ASYNC_PER_TILE in flight
}
__device__ __forceinline__ void wait_async_0() {
  asm volatile("s_wait_asynccnt 0x0" ::: "memory");
}

// ---------------------------------------------------------------------------
// WMMA fp32 GEMV core: out[rowBase .. rowBase+63] = W[rows,:] @ xin (+ addv)
//
// Per ISA 7.12.2 (32-bit A 16x4): lanes 0-15 hold {K=0,K=1}, lanes 16-31 hold
// {K=2,K=3} for rows M=lane%16.  B 4x16 mirrors that K split per VGPR; we
// broadcast xin[k] across all 16 columns so every column of C is the same
// dot product.  C 16x16 f32: VGPR j = row j (lanes 0-15) / row j+8 (lanes
// 16-31), so lane 0 and lane 16 own the final per-row sums.
//
// Double-buffered: async-copy tile i+1 while WMMAs consume tile i.
// ---------------------------------------------------------------------------
__device__ __forceinline__ void gemv_wmma_core(const float* __restrict__ W,
                                               const float* __restrict__ xin,
                                               float* __restrict__ out,
                                               const float* __restrict__ addv,
                                               int rowTileBase,
                                               float* xs,                  // [D]
                                               float (*tileA)[RPB][KTP]) { // [2][RPB][KTP]
  const int t = threadIdx.x;
  const int wave = t >> 5;
  const int lane = t & 31;
  const bool hi = lane >= 16;
  const int mrow = wave * 16 + (lane & 15);   // LDS row for this lane's A frag

  // Stage x into LDS once (coalesced float4; uniform counted loop, no masking)
#pragma unroll 4
  for (int j = 0; j < (D / 4) / TPB; ++j) {
    const int i = t + j * TPB;
    ((float4*)xs)[i] = ((const float4*)xin)[i];
  }

  // Prefetch tile 0
  stage_tile_async(W, rowTileBase, 0, tileA[0]);

  v8f c = {0.f, 0.f, 0.f, 0.f, 0.f, 0.f, 0.f, 0.f};

  for (int it = 0; it < NTILES; ++it) {
    const int cur = it & 1;

    // Issue async copy for the next tile into the other buffer (safe: the
    // trailing barrier of iteration it-1 closed all compute on that buffer),
    // then wait so that only the just-issued copies remain in flight.
    if (it + 1 < NTILES) {
      stage_tile_async(W, rowTileBase, (it + 1) * KT, tileA[1 - cur]);
      wait_async_n();
    } else {
      wait_async_0();
    }
    __syncthreads();                // all waves' current-tile data visible

    const float(*tile)[KTP] = tileA[cur];
    const int kbase = it * KT;
#pragma unroll
    for (int kk = 0; kk < KT; kk += 4) {
      v2f a, b;
      const int acol = kk + (hi ? 2 : 0);
      a.x = tile[mrow][acol];
      a.y = tile[mrow][acol + 1];
      b.x = xs[kbase + kk + (hi ? 2 : 0)];
      b.y = xs[kbase + kk + (hi ? 3 : 1)];
      c = __builtin_amdgcn_wmma_f32_16x16x4_f32(
          /*neg_a=*/false, a, /*neg_b=*/false, b,
          /*c_mod=*/(short)0, c, /*reuse_a=*/false, /*reuse_b=*/false);
    }
    __syncthreads();                // close compute before buffer is overwritten
  }

  // Write out: lane 0 owns rows [base..base+7], lane 16 owns [base+8..base+15]
  if ((lane & 15) == 0) {
    const int mbase = rowTileBase + wave * 16 + (hi ? 8 : 0);
#pragma unroll
    for (int j = 0; j < 8; ++j) {
      float val = c[j];
      if (addv) val += addv[mbase + j];
      out[mbase + j] = val;
    }
  }
}

// Fused r/k/v GEMV: grid = 3 * (D/RPB) blocks
__global__ void gemv3_wmma_kernel(const float* __restrict__ Wr,
                                  const float* __restrict__ Wk,
                                  const float* __restrict__ Wv,
                                  const float* __restrict__ x,
                                  float* __restrict__ outr,
                                  float* __restrict__ outk,
                                  float* __restrict__ outv) {
  __shared__ float xs[D];
  __shared__ float tileA[2][RPB][KTP];
  const int TILES = D / RPB;
  const int mat = blockIdx.x / TILES;
  const int tile = blockIdx.x % TILES;
  const float* W = (mat == 0) ? Wr : (mat == 1) ? Wk : Wv;
  float* out = (mat == 0) ? outr : (mat == 1) ? outk : outv;
  gemv_wmma_core(W, x, out, nullptr, tile * RPB, xs, tileA);
}

// h = x + Wo @ y : grid = D/RPB blocks
__global__ void gemv_add_wmma_kernel(const float* __restrict__ Wo,
                                     const float* __restrict__ y,
                                     const float* __restrict__ x,
                                     float* __restrict__ h) {
  __shared__ float xs[D];
  __shared__ float tileA[2][RPB][KTP];
  gemv_wmma_core(Wo, y, h, x, blockIdx.x * RPB, xs, tileA);
}

// ---------------------------------------------------------------------------
// Kernel 3: elementwise Kalman/RoPE mix -> y
// ---------------------------------------------------------------------------
__device__ __forceinline__ float softplusf(float x) {
  return (x > 20.f) ? x : log1pf(expf(x));
}
__device__ __forceinline__ float clampf(float x, float lo, float hi) {
  return fminf(fmaxf(x, lo), hi);
}

__global__ void mix_kernel(const float* __restrict__ rpre,
                           const float* __restrict__ kvec,
                           const float* __restrict__ vvec,
                           const float* __restrict__ state_num,
                           const float* __restrict__ state_den,
                           const float* __restrict__ state_var,
                           const float* __restrict__ raw_decay,
                           const float* __restrict__ pn_param,
                           const float* __restrict__ on_param,
                           const float* __restrict__ step_pos,
                           const float* __restrict__ W_time,
                           float* __restrict__ y) {
  const int i = blockIdx.x * blockDim.x + threadIdx.x;
  if (i >= D) return;

  float v = vvec[i];
  if (i < 2 * M_TIME) {               // RoPE on first 16 pairs
    const int p = i >> 1;
    const float th = step_pos[0] * W_time[p];
    const float cth = cosf(th), sth = sinf(th);
    const float v0 = vvec[2 * p], v1 = vvec[2 * p + 1];
    v = (i & 1) ? (v0 * sth + v1 * cth) : (v0 * cth - v1 * sth);
  }

  const float rate = fmaxf(-softplusf(raw_decay[i]), -MAX_EXP);
  const float lam = fmaxf(expf(rate), EPS_DIV);
  const float pn = clampf(softplusf(pn_param[0]), EPS_DIV, MAX_NOISE);
  const float on = clampf(softplusf(on_param[0]), EPS_DIV, MAX_NOISE);

  const float sn = state_num[i] * lam;
  const float sd = state_den[i] * lam;
  const float sv = state_var[i] * lam * lam + pn;

  const float k = kvec[i];
  const float r = 1.f / (1.f + expf(-rpre[i]));
  const float w = expf(fminf(k, MAX_EXP));
  const float pred_state = sn / (sd + EPS_DIV);
  const float resid = w * v - pred_state;
  const float obs_var = expf(fminf(-k, MAX_EXP)) * on + EPS_DIV;
  const float gain = clampf(sv / (sv + obs_var), EPS_DIV, 1.f - EPS_DIV);
  const float new_state = pred_state + gain * resid;
  y[i] = r * new_state;
}

// ---------------------------------------------------------------------------
// Kernel 5: exact k-th largest via 32-pass bitwise radix select, then
// threshold (h >= thresh keeps ties, matching the reference).  One block.
// ---------------------------------------------------------------------------
__global__ void kwta_kernel(const float* __restrict__ h, float* __restrict__ out) {
  const int t = threadIdx.x;               // 1024 threads
  const int ELTS = D / 1024;               // 8 per thread
  __shared__ unsigned s_cnt;

  float hv[ELTS];
  unsigned u[ELTS];
#pragma unroll
  for (int j = 0; j < ELTS; ++j) {
    const float f = h[j * 1024 + t];
    hv[j] = f;
    unsigned b = __float_as_uint(f);
    u[j] = (b & 0x80000000u) ? ~b : (b | 0x80000000u);  // order-preserving map
  }

  unsigned prefix = 0;
  int remaining = KWTA_K;
  for (int bit = 31; bit >= 0; --bit) {
    if (t == 0) s_cnt = 0;
    __syncthreads();
    const unsigned testPrefix = prefix | (1u << bit);
    const unsigned mask = ~((1u << bit) - 1u);   // bits [31:bit]
    unsigned local = 0;
#pragma unroll
    for (int j = 0; j < ELTS; ++j) {
      if ((u[j] & mask) == testPrefix) ++local;
    }
    if (local) atomicAdd(&s_cnt, local);
    __syncthreads();
    const int c1 = (int)s_cnt;
    if (c1 >= remaining) {
      prefix = testPrefix;
    } else {
      remaining -= c1;
    }
    __syncthreads();
  }

  // prefix == ordered key of the K-th largest element; u >= prefix <=> h >= thresh
#pragma unroll
  for (int j = 0; j < ELTS; ++j) {
    out[j * 1024 + t] = (u[j] >= prefix) ? hv[j] : 0.f;
  }
}

// ---------------------------------------------------------------------------
// Launch
// ---------------------------------------------------------------------------
extern "C" void kernel_launch(void* const* d_in, const int* in_sizes, int n_in,
                              void* d_out, int out_size, void* d_ws, size_t ws_size,
                              hipStream_t stream) {
  const float* x_in      = (const float*)d_in[0];
  const float* step_pos  = (const float*)d_in[1];
  const float* pred      = (const float*)d_in[2];
  const float* state_num = (const float*)d_in[3];
  const float* state_den = (const float*)d_in[4];
  const float* state_var = (const float*)d_in[5];
  const float* Wr        = (const float*)d_in[6];
  const float* Wk        = (const float*)d_in[7];
  const float* Wv        = (const float*)d_in[8];
  const float* Wo        = (const float*)d_in[9];
  const float* raw_decay = (const float*)d_in[10];
  const float* pn_param  = (const float*)d_in[11];
  const float* on_param  = (const float*)d_in[12];
  const float* g_norm    = (const float*)d_in[13];
  const float* W_time    = (const float*)d_in[14];

  float* ws   = (float*)d_ws;
  float* x    = ws + 0 * D;
  float* rpre = ws + 1 * D;
  float* kv   = ws + 2 * D;
  float* vv   = ws + 3 * D;
  float* y    = ws + 4 * D;
  float* h    = ws + 5 * D;

  rmsnorm_kernel<<<1, 256, 0, stream>>>(x_in, pred, g_norm, x);

  gemv3_wmma_kernel<<<3 * (D / RPB), TPB, 0, stream>>>(Wr, Wk, Wv, x, rpre, kv, vv);

  mix_kernel<<<D / 256, 256, 0, stream>>>(rpre, kv, vv, state_num, state_den,
                                          state_var, raw_decay, pn_param,
                                          on_param, step_pos, W_time, y);

  gemv_add_wmma_kernel<<<D / RPB, TPB, 0, stream>>>(Wo, y, x, h);

  kwta_kernel<<<1, 1024, 0, stream>>>(h, (float*)d_out);
}